// GunGNN_69380901700243
// MI455X (gfx1250) — compile-verified
//
#include <hip/hip_runtime.h>

#define N_NODES 100000
#define N_EDGES 1600000
#define N_GRAPHS 64
#define CH 32
#define FC_OUT 8

typedef float v2f __attribute__((ext_vector_type(2)));
typedef float v8f __attribute__((ext_vector_type(8)));

// ---------------- degree / normalization ----------------
__global__ void k_init_deg(float* __restrict__ deg) {
  int n = blockIdx.x * 256 + threadIdx.x;
  if (n < N_NODES) deg[n] = 1.0f;   // self-loop contributes 1
}

__global__ void k_zero4(float4* __restrict__ p, int n4) {
  int t = blockIdx.x * 256 + threadIdx.x;
  if (t < n4) p[t] = make_float4(0.f, 0.f, 0.f, 0.f);
}

__global__ void k_degree(const int* __restrict__ ei, float* __restrict__ deg) {
  int e = blockIdx.x * 256 + threadIdx.x;
  if (e < N_EDGES) {
    int d = ei[2 * (N_EDGES + e)];   // low dword of int64 dst index
    atomicAdd(deg + d, 1.0f);
  }
}

__global__ void k_rsqrt_inplace(float* __restrict__ deg) {
  int n = blockIdx.x * 256 + threadIdx.x;
  if (n < N_NODES) deg[n] = rsqrtf(deg[n]);
}

// ---------------- fp32 WMMA GEMM: H[nrows,32] = X[nrows,32] @ W[32,32] ----------------
// One wave32 per 16-row block. D(16x16) = A(16x4) x B(4x16) + C, 8 K-steps, 2 N-tiles.
// A layout: lanes 0-15 hold M=0..15 with K={0,1}; lanes 16-31 hold K={2,3}.
// B layout: VGPR0 = K-row (0 | 2) across lanes (N=lane&15), VGPR1 = K-row (1 | 3).
// C/D layout: VGPR v -> M = v (lanes 0-15) / v+8 (lanes 16-31), N = lane&15.
__global__ void k_gemm32_wmma(const float* __restrict__ X, const float* __restrict__ W,
                              float* __restrict__ H, int nrows) {
  const int wave = (blockIdx.x * blockDim.x + threadIdx.x) >> 5;
  const int lane = threadIdx.x & 31;
  const int row0 = wave << 4;
  if (row0 >= nrows) return;            // wave-uniform: EXEC all-ones inside

  const int m    = lane & 15;
  const int koff = (lane >> 4) << 1;    // 0 or 2

  v8f acc0 = {};
  v8f acc1 = {};
  const float* __restrict__ xr = X + (row0 + m) * CH;

#pragma unroll
  for (int kb = 0; kb < CH; kb += 4) {
    v2f a, b0, b1;
    a.x = xr[kb + koff];
    a.y = xr[kb + koff + 1];
    const float* __restrict__ w0 = W + (kb + koff) * CH;   // K row (kb+koff)
    b0.x = w0[m];           b0.y = w0[CH + m];             // N-tile 0
    b1.x = w0[16 + m];      b1.y = w0[CH + 16 + m];        // N-tile 1
    acc0 = __builtin_amdgcn_wmma_f32_16x16x4_f32(false, a, false, b0,
                                                 (short)0, acc0, false, false);
    acc1 = __builtin_amdgcn_wmma_f32_16x16x4_f32(false, a, false, b1,
                                                 (short)0, acc1, false, false);
  }

  const int half = lane >> 4;
  const int col  = lane & 15;
#pragma unroll
  for (int v = 0; v < 8; v++) {
    int row = row0 + (half << 3) + v;
    H[row * CH + col]      = acc0[v];
    H[row * CH + 16 + col] = acc1[v];
  }
}

// ---------------- edge scatter: AGG[dst] += H[src] * dinv[src]*dinv[dst] ----------------
// 4 threads per edge, 8 channels each (two float4 gathers, 8 L2-resident f32 atomics).
__global__ void k_edge_agg(const float* __restrict__ H, const float* __restrict__ dinv,
                           const int* __restrict__ ei, float* __restrict__ AGG) {
  int t = blockIdx.x * 256 + threadIdx.x;
  int e = t >> 2;
  if (e >= N_EDGES) return;
  int part = (t & 3) << 3;
  int s = ei[2 * e];
  int d = ei[2 * (N_EDGES + e)];
  float nrm = dinv[s] * dinv[d];
  const float4* __restrict__ hs = (const float4*)(H + s * CH + part);
  float4 v0 = hs[0];
  float4 v1 = hs[1];
  float* o = AGG + d * CH + part;
  atomicAdd(o + 0, v0.x * nrm); atomicAdd(o + 1, v0.y * nrm);
  atomicAdd(o + 2, v0.z * nrm); atomicAdd(o + 3, v0.w * nrm);
  atomicAdd(o + 4, v1.x * nrm); atomicAdd(o + 5, v1.y * nrm);
  atomicAdd(o + 6, v1.z * nrm); atomicAdd(o + 7, v1.w * nrm);
}

// ---------------- self-loop + bias + relu, in place on AGG ----------------
__global__ void k_finalize(float* __restrict__ AGG, const float* __restrict__ H,
                           const float* __restrict__ dinv, const float* __restrict__ bias) {
  int t = blockIdx.x * 256 + threadIdx.x;
  if (t >= N_NODES * CH) return;
  int n = t >> 5, c = t & 31;
  float di = dinv[n];
  float v = AGG[t] + H[t] * (di * di) + bias[c];
  AGG[t] = fmaxf(v, 0.0f);
}

// ---------------- global mean pool (sum + count via atomics) ----------------
__global__ void k_pool(const float* __restrict__ ACT, const int* __restrict__ bat,
                       float* __restrict__ pool, float* __restrict__ cnt) {
  int t = blockIdx.x * 256 + threadIdx.x;
  if (t >= N_NODES * CH) return;
  int n = t >> 5, c = t & 31;
  int g = bat[2 * n];                  // low dword of int64 batch id
  atomicAdd(pool + g * CH + c, ACT[t]);
  if (c == 0) atomicAdd(cnt + g, 1.0f);
}

// ---------------- final FC: out[64,8] = (pool/cnt) @ Wfc + bfc ----------------
__global__ void k_fc(const float* __restrict__ pool, const float* __restrict__ cnt,
                     const float* __restrict__ Wfc, const float* __restrict__ bfc,
                     float* __restrict__ out) {
  int t = threadIdx.x;                 // 512 threads, one per output
  int g = t >> 3, o = t & 7;
  float inv = 1.0f / fmaxf(cnt[g], 1.0f);
  float acc = bfc[o];
#pragma unroll
  for (int c = 0; c < CH; c++)
    acc += (pool[g * CH + c] * inv) * Wfc[c * FC_OUT + o];
  out[t] = acc;
}

extern "C" void kernel_launch(void* const* d_in, const int* in_sizes, int n_in,
                              void* d_out, int out_size, void* d_ws, size_t ws_size,
                              hipStream_t stream) {
  const float* x   = (const float*)d_in[0];
  const int*   ei  = (const int*)d_in[1];   // int64 [2,E]; values < 2^31 -> read low dwords
  const int*   bat = (const int*)d_in[2];   // int64 [N]
  const float* W1  = (const float*)d_in[3];
  const float* b1  = (const float*)d_in[4];
  const float* W2  = (const float*)d_in[5];
  const float* b2  = (const float*)d_in[6];
  const float* Wfc = (const float*)d_in[7];
  const float* bfc = (const float*)d_in[8];
  float* out = (float*)d_out;

  // workspace layout (floats, 64-float aligned offsets)
  float* ws   = (float*)d_ws;
  float* dinv = ws;                          // 100000 (deg -> dinv in place)
  float* H    = ws + 100352;                 // 3,200,000
  float* AGG  = H + N_NODES * CH;            // 3,200,000
  float* POOL = AGG + N_NODES * CH;          // 2048
  float* CNT  = POOL + N_GRAPHS * CH;        // 64

  const int gN    = (N_NODES + 255) / 256;
  const int gE    = (N_EDGES + 255) / 256;
  const int gE4   = (4 * N_EDGES + 255) / 256;
  const int gNC   = (N_NODES * CH + 255) / 256;
  const int gNC4  = (N_NODES * CH / 4 + 255) / 256;
  const int gGemm = ((N_NODES / 16) * 32 + 255) / 256;   // one wave32 per 16 rows

  // degrees -> dinv
  k_init_deg<<<gN, 256, 0, stream>>>(dinv);
  k_zero4<<<(2112 / 4 + 255) / 256, 256, 0, stream>>>((float4*)POOL, 2112 / 4);
  k_degree<<<gE, 256, 0, stream>>>(ei, dinv);
  k_rsqrt_inplace<<<gN, 256, 0, stream>>>(dinv);

  // layer 1
  k_gemm32_wmma<<<gGemm, 256, 0, stream>>>(x, W1, H, N_NODES);
  k_zero4<<<gNC4, 256, 0, stream>>>((float4*)AGG, N_NODES * CH / 4);
  k_edge_agg<<<gE4, 256, 0, stream>>>(H, dinv, ei, AGG);
  k_finalize<<<gNC, 256, 0, stream>>>(AGG, H, dinv, b1);   // AGG = act1

  // layer 2 (reuse H for h2, then AGG for agg2/act2)
  k_gemm32_wmma<<<gGemm, 256, 0, stream>>>(AGG, W2, H, N_NODES);
  k_zero4<<<gNC4, 256, 0, stream>>>((float4*)AGG, N_NODES * CH / 4);
  k_edge_agg<<<gE4, 256, 0, stream>>>(H, dinv, ei, AGG);
  k_finalize<<<gNC, 256, 0, stream>>>(AGG, H, dinv, b2);   // AGG = act2

  // pool + FC head
  k_pool<<<gNC, 256, 0, stream>>>(AGG, bat, POOL, CNT);
  k_fc<<<1, N_GRAPHS * FC_OUT, 0, stream>>>(POOL, CNT, Wfc, bfc, out);
}